// GRUCellWrapper_67242007986844
// MI455X (gfx1250) — compile-verified
//
#include <hip/hip_runtime.h>
#include <hip/hip_bf16.h>

typedef __attribute__((ext_vector_type(2))) float v2f;
typedef __attribute__((ext_vector_type(8))) float v8f;

#define Bq 128
#define Tq 1024
#define Iq 256
#define Hq 512
#define NWG 32            // each WG owns 16 H-columns
#define PH (Hq + 4)       // LDS pitch for W_hh rows (bank-conflict free)
#define PI (Iq + 4)       // LDS pitch for W_ih rows

__device__ __forceinline__ float sigmoidf_(float x) {
  return 1.0f / (1.0f + __expf(-x));
}
__device__ __forceinline__ float tanhf_(float x) {
  x = fminf(fmaxf(x, -15.0f), 15.0f);
  float e = __expf(-2.0f * x);
  return (1.0f - e) / (1.0f + e);
}

__global__ void gru_init_kernel(unsigned* cnt) {
  int i = blockIdx.x * blockDim.x + threadIdx.x;
  if (i < Tq) cnt[i] = 0u;
}

__global__ __launch_bounds__(256, 1)
void gru_scan_kernel(const float* __restrict__ x,     // (B,T,I)
                     const float* __restrict__ h0,    // (B,H)
                     const float* __restrict__ Wih,   // (3H,I)
                     const float* __restrict__ bih,   // (3H)
                     const float* __restrict__ Whh,   // (3H,H)
                     const float* __restrict__ bhh,   // (3H)
                     float* __restrict__ out,         // 2 x (B,4H)
                     float* __restrict__ hbuf,        // [2][B*H] scratch
                     unsigned* __restrict__ cnt)      // [T] barrier counters
{
  // Per-WG weight slice resident in LDS for the whole scan (CDNA5: 320KB/WGP)
  __shared__ float sWhh[48 * PH];   // rows: {r,z,n} x 16 cols, K=512
  __shared__ float sWih[48 * PI];   // rows: {r,z,n} x 16 cols, K=256

  const int wg   = blockIdx.x;      // 0..31
  const int c0   = wg * 16;         // base H column of this WG
  const int tid  = threadIdx.x;
  const int wid  = tid >> 5;        // wave id == M-tile (0..7)
  const int lane = tid & 31;
  const int ln   = lane & 15;
  const int lh   = lane >> 4;       // half-wave selects K pair / row group

  // Stage W_hh rows {g, 512+g, 1024+g : g in [c0, c0+16)} -> LDS
  for (int idx = tid; idx < 48 * Hq; idx += 256) {
    int r0 = idx / Hq, k = idx % Hq;
    int g  = (r0 >> 4) * Hq + c0 + (r0 & 15);
    sWhh[r0 * PH + k] = Whh[g * Hq + k];
  }
  for (int idx = tid; idx < 48 * Iq; idx += 256) {
    int r0 = idx / Iq, k = idx % Iq;
    int g  = (r0 >> 4) * Hq + c0 + (r0 & 15);
    sWih[r0 * PI + k] = Wih[g * Iq + k];
  }
  __syncthreads();

  // Per-lane output column + time-invariant biases
  const int   hj  = c0 + ln;
  const float rb  = bih[hj]          + bhh[hj];
  const float zb  = bih[Hq + hj]     + bhh[Hq + hj];
  const float nbi = bih[2 * Hq + hj];
  const float nbh = bhh[2 * Hq + hj];

  const int    arow = wid * 16 + ln;                 // A-fragment row
  const float* xrow = x + (size_t)arow * (Tq * Iq);

  for (int t = 0; t < Tq; ++t) {
    v8f accR = {}, accZ = {}, accNH = {}, accNI = {};

    // ---- input projection first: K = 256 over x[:, t, :] ----
    // Independent of h_{t-1}; runs while other WGs finish step t-1,
    // hiding the inter-step barrier latency behind 1/3 of the GEMM.
    const float* xa = xrow + t * Iq;
    #pragma unroll 4
    for (int kk = 0; kk < Iq; kk += 4) {
      const int kb = kk + 2 * lh;
      v2f a  = *(const v2f*)(xa + kb);                           // global_load_b64
      v2f bR = *(const v2f*)(&sWih[( 0 + ln) * PI + kb]);        // ds_load_b64
      v2f bZ = *(const v2f*)(&sWih[(16 + ln) * PI + kb]);
      v2f bN = *(const v2f*)(&sWih[(32 + ln) * PI + kb]);
      accR  = __builtin_amdgcn_wmma_f32_16x16x4_f32(false, a, false, bR, (short)0, accR,  false, false);
      accZ  = __builtin_amdgcn_wmma_f32_16x16x4_f32(false, a, false, bZ, (short)0, accZ,  false, false);
      accNI = __builtin_amdgcn_wmma_f32_16x16x4_f32(false, a, false, bN, (short)0, accNI, false, false);
    }

    // Prefetch next step's x A-fragment lines (global_prefetch_b8)
    if (t + 1 < Tq) {
      __builtin_prefetch(xa + Iq + 2 * lh, 0, 0);
      __builtin_prefetch(xa + Iq + 128 + 2 * lh, 0, 0);
    }

    // ---- wait until h_{t-1} is complete (split barrier: wait side) ----
    if (t > 0) {
      if (tid == 0) {
        while (__hip_atomic_load(&cnt[t - 1], __ATOMIC_ACQUIRE,
                                 __HIP_MEMORY_SCOPE_AGENT) < (unsigned)NWG) {
          __builtin_amdgcn_s_sleep(1);
        }
      }
      __syncthreads();
    }

    const float* hsrc = (t == 0) ? h0 : (hbuf + (size_t)((t - 1) & 1) * (Bq * Hq));
    float*       hdst = hbuf + (size_t)(t & 1) * (Bq * Hq);

    // ---- recurrent contribution: K = 512 over h_{t-1} ----
    const float* ha = hsrc + (size_t)arow * Hq;
    #pragma unroll 4
    for (int kk = 0; kk < Hq; kk += 4) {
      const int kb = kk + 2 * lh;
      v2f a  = *(const v2f*)(ha + kb);
      v2f bR = *(const v2f*)(&sWhh[( 0 + ln) * PH + kb]);
      v2f bZ = *(const v2f*)(&sWhh[(16 + ln) * PH + kb]);
      v2f bN = *(const v2f*)(&sWhh[(32 + ln) * PH + kb]);
      accR  = __builtin_amdgcn_wmma_f32_16x16x4_f32(false, a, false, bR, (short)0, accR,  false, false);
      accZ  = __builtin_amdgcn_wmma_f32_16x16x4_f32(false, a, false, bZ, (short)0, accZ,  false, false);
      accNH = __builtin_amdgcn_wmma_f32_16x16x4_f32(false, a, false, bN, (short)0, accNH, false, false);
    }

    // ---- gates + hidden update (C/D layout: lane col = ln, row = v + 8*lh) ----
    #pragma unroll
    for (int v = 0; v < 8; ++v) {
      const int m  = wid * 16 + 8 * lh + v;
      float r  = sigmoidf_(accR[v] + rb);
      float z  = sigmoidf_(accZ[v] + zb);
      float nn = tanhf_(accNI[v] + nbi + r * (accNH[v] + nbh));
      float ho = hsrc[(size_t)m * Hq + hj];
      float hn = (1.0f - z) * nn + z * ho;
      hdst[(size_t)m * Hq + hj] = hn;
      if (t == Tq - 1) {
        float* o0 = out + (size_t)m * (4 * Hq);
        o0[hj]          = r;
        o0[Hq + hj]     = z;
        o0[2 * Hq + hj] = nn;
        o0[3 * Hq + hj] = hn;
        float* o1 = o0 + (size_t)Bq * (4 * Hq);
        o1[hj]          = r;
        o1[Hq + hj]     = z;
        o1[2 * Hq + hj] = nn;
        o1[3 * Hq + hj] = hn;
      }
    }

    // ---- split barrier: signal side (h(t) fully written by this WG) ----
    __threadfence();
    __syncthreads();
    if (tid == 0 && t + 1 < Tq) {
      __hip_atomic_fetch_add(&cnt[t], 1u, __ATOMIC_ACQ_REL, __HIP_MEMORY_SCOPE_AGENT);
    }
  }
}

extern "C" void kernel_launch(void* const* d_in, const int* in_sizes, int n_in,
                              void* d_out, int out_size, void* d_ws, size_t ws_size,
                              hipStream_t stream) {
  (void)in_sizes; (void)n_in; (void)out_size; (void)ws_size;
  const float* x   = (const float*)d_in[0];
  const float* h   = (const float*)d_in[1];
  const float* Wih = (const float*)d_in[2];
  const float* bih = (const float*)d_in[3];
  const float* Whh = (const float*)d_in[4];
  const float* bhh = (const float*)d_in[5];
  float* out = (float*)d_out;

  float*    hbuf = (float*)d_ws;                                        // 512 KB
  unsigned* cnt  = (unsigned*)((char*)d_ws + 2ull * Bq * Hq * sizeof(float));

  gru_init_kernel<<<4, 256, 0, stream>>>(cnt);
  gru_scan_kernel<<<NWG, 256, 0, stream>>>(x, h, Wih, bih, Whh, bhh, out, hbuf, cnt);
}